// CBDC_loss_22849226014778
// MI455X (gfx1250) — compile-verified
//
#include <hip/hip_runtime.h>
#include <stdint.h>

#define Bb 2
#define Dd 64
#define Hh 96
#define Ww 96
#define HW (Hh*Ww)            // 9216
#define VOL (Dd*HW)           // 589824
#define NVOL 4                // [pred b0, pred b1, true b0, true b1]
#define NV (NVOL*VOL)         // 2359296
#define NBATCH (Bb*VOL)       // 1179648
#define INF_D2 1.0e12f
#define EPSF 1e-07f

typedef __attribute__((ext_vector_type(2))) float v2f;
typedef __attribute__((ext_vector_type(8))) float v8f;

// ---------------------------------------------------------------- binarize
__global__ void k_binarize(const float* __restrict__ yp, const int* __restrict__ yt,
                           float* __restrict__ mask) {
    int i = blockIdx.x * blockDim.x + threadIdx.x;
    if (i >= NBATCH) return;
    int b = i / VOL, r = i % VOL;
    float l0 = yp[(b * 2 + 0) * VOL + r];
    float l1 = yp[(b * 2 + 1) * VOL + r];           // fore = max over channels 1: (C==2)
    mask[b * VOL + r]       = (l1 > l0) ? 1.0f : 0.0f;   // prob1 > 0.5  <=>  l1 > l0
    mask[(2 + b) * VOL + r] = (yt[b * VOL + r] > 0) ? 1.0f : 0.0f;
}

// ---------------------------------------------------- soft erode (7-pt cross min)
__global__ void k_erode(const float* __restrict__ in, float* __restrict__ out) {
    int i = blockIdx.x * blockDim.x + threadIdx.x;
    if (i >= NV) return;
    int v = i / VOL, r = i % VOL;
    int d = r / HW, r2 = r % HW, h = r2 / Ww, w = r2 % Ww;
    const float* p = in + v * VOL;
    float m = p[r];
    if (d > 0)      m = fminf(m, p[r - HW]);
    if (d < Dd - 1) m = fminf(m, p[r + HW]);
    if (h > 0)      m = fminf(m, p[r - Ww]);
    if (h < Hh - 1) m = fminf(m, p[r + Ww]);
    if (w > 0)      m = fminf(m, p[r - 1]);
    if (w < Ww - 1) m = fminf(m, p[r + 1]);
    out[i] = m;
}

// ------------------------- fused 3x3x3 dilate of e + skeleton init/update
__global__ void k_dilate_update(const float* __restrict__ e, const float* __restrict__ x,
                                float* __restrict__ skel, int is_init) {
    int i = blockIdx.x * blockDim.x + threadIdx.x;
    if (i >= NV) return;
    int v = i / VOL, r = i % VOL;
    int d = r / HW, r2 = r % HW, h = r2 / Ww, w = r2 % Ww;
    int d0 = d > 0 ? d - 1 : 0, d1 = d < Dd - 1 ? d + 1 : Dd - 1;
    int h0 = h > 0 ? h - 1 : 0, h1 = h < Hh - 1 ? h + 1 : Hh - 1;
    int w0 = w > 0 ? w - 1 : 0, w1 = w < Ww - 1 ? w + 1 : Ww - 1;
    const float* p = e + v * VOL;
    float mx = -3.4e38f;
    for (int dz = d0; dz <= d1; ++dz)
        for (int dy = h0; dy <= h1; ++dy) {
            int base = dz * HW + dy * Ww;
            for (int dx = w0; dx <= w1; ++dx)
                mx = fmaxf(mx, p[base + dx]);
        }
    float delta = fmaxf(x[i] - mx, 0.0f);        // relu(x - open)
    if (is_init) {
        skel[i] = delta;
    } else {
        float s = skel[i];
        skel[i] = s + fmaxf(delta - s * delta, 0.0f);
    }
}

// ---------------------------------------------------------------- EDT init
__global__ void k_edt_init(const float* __restrict__ mask, float* __restrict__ d2) {
    int i = blockIdx.x * blockDim.x + threadIdx.x;
    if (i >= NV) return;
    d2[i] = (mask[i] > 0.0f) ? INF_D2 : 0.0f;
}

// ----------- EDT 1D pass: out[i] = min_j (i-j)^2 + d2[j]; line staged to LDS
// via the CDNA5 async-to-LDS path (ASYNCcnt-tracked DMA into shared memory).
__global__ void k_edt_pass(float* __restrict__ d2, int L, int stride,
                           int n1, int s1, int n2, int s2) {
    __shared__ float line[96];
    int lid = blockIdx.x;
    int v = lid / (n1 * n2);
    int rem = lid % (n1 * n2);
    int i1 = rem / n2, i2 = rem % n2;
    int base = v * VOL + i1 * s1 + i2 * s2;
    int t = threadIdx.x;                         // 0..L-1, blockDim.x == L
    float* gp = d2 + base + t * stride;

    unsigned lds_addr = (unsigned)(uintptr_t)(&line[t]);   // low 32 bits = LDS offset
    asm volatile(
        "global_load_async_to_lds_b32 %0, %1, off\n\t"
        "s_wait_asynccnt 0"
        :: "v"(lds_addr), "v"(gp) : "memory");
    __syncthreads();

    float ti = (float)t;
    float best = 3.4e38f;
    for (int j = 0; j < L; ++j) {
        float dj = ti - (float)j;                // |dj| <= 95 -> dj*dj exact in fp32
        best = fminf(best, __builtin_fmaf(dj, dj, line[j]));
    }
    *gp = best;                                  // LDS holds the copy; safe in-place
}

// ------------------------------------------------------------- rmax per volume
__global__ void k_rmax_init(float* rmax) {
    int i = threadIdx.x;
    if (i < NVOL) rmax[i] = 0.0f;
}

__global__ void k_rmax(const float* __restrict__ mask, const float* __restrict__ skel,
                       const float* __restrict__ d2, float* __restrict__ rmax) {
    const int CHUNK = 4096;                      // 144 blocks per volume * 4096 = VOL
    int v = blockIdx.x / 144;
    int blk = blockIdx.x % 144;
    int base = v * VOL + blk * CHUNK;
    float m = 0.0f;
    for (int t = threadIdx.x; t < CHUNK; t += blockDim.x) {
        int i = base + t;
        float dist = (mask[i] == 0.0f) ? 0.0f : sqrtf(d2[i]);
        float val = (skel[i] == 1.0f) ? dist : 0.0f;
        m = fmaxf(m, val);
    }
    __shared__ float red[256];
    red[threadIdx.x] = m;
    __syncthreads();
    for (int s = 128; s > 0; s >>= 1) {
        if (threadIdx.x < s) red[threadIdx.x] = fmaxf(red[threadIdx.x], red[threadIdx.x + s]);
        __syncthreads();
    }
    if (threadIdx.x == 0)
        atomicMax((unsigned int*)&rmax[v], __float_as_uint(red[0]));  // vals >= 0
}

// ---------------- fused weights + four global sums via V_WMMA_F32_16X16X4_F32
// Each wave feeds 64 per-voxel terms through the matrix pipe (A 16x4, B = ones).
__global__ void k_sums_wmma(const float* __restrict__ mask, const float* __restrict__ skel,
                            const float* __restrict__ d2, const float* __restrict__ rmax,
                            float* __restrict__ partials) {
    const int GROUPS = NBATCH / 64;              // 18432, divides evenly by total waves
    int wave = (blockIdx.x * blockDim.x + threadIdx.x) >> 5;
    int lane = threadIdx.x & 31;
    int nwaves = (gridDim.x * blockDim.x) >> 5;  // 2048 with <<<256,256>>>

    v8f a1 = {0.f,0.f,0.f,0.f,0.f,0.f,0.f,0.f};
    v8f a2 = a1, a3 = a1, a4 = a1;
    v2f bones = {1.0f, 1.0f};

    for (int g = wave; g < GROUPS; g += nwaves) {
        int vbase = g * 64 + lane * 2;
        v2f t1, t2, t3, t4;
        for (int u = 0; u < 2; ++u) {
            int i = vbase + u;                   // index into [B,D,H,W]
            int b = i / VOL, r = i % VOL;
            int ip = b * VOL + r;                // pred volume
            int it = (2 + b) * VOL + r;          // true volume
            float rp = rmax[b], rt = rmax[2 + b];
            float rsp = (rp > 0.0f) ? rp : 1.0f;
            float rst = (rt > 0.0f) ? rt : 1.0f;
            float distp = (mask[ip] == 0.0f) ? 0.0f : sqrtf(d2[ip]);
            float distt = (mask[it] == 0.0f) ? 0.0f : sqrtf(d2[it]);
            float q_vp = distp / rsp;            // dist_n (pred)
            float q_vl = distt / rst;            // dist_n (true)
            bool onp = (skel[ip] == 1.0f);
            bool ont = (skel[it] == 1.0f);
            float q_spvp = onp ? q_vp : 0.0f;    // skel_rn (pred)
            float q_slvl = ont ? q_vl : 0.0f;    // skel_rn (true)
            float q_sp = onp ? (1.0f + EPSF) / (q_spvp * q_spvp + EPSF) : 0.0f;
            float q_sl = ont ? (1.0f + EPSF) / (q_slvl * q_slvl + EPSF) : 0.0f;
            float c1 = q_sp * q_vl;
            float c2 = ((q_spvp != 0.0f) && (q_slvl == 0.0f)) ? q_spvp * q_sp : q_slvl * q_sp;
            float c3 = q_sl * q_vp;
            float c4 = ((q_slvl != 0.0f) && (q_spvp == 0.0f)) ? q_slvl * q_sl : q_spvp * q_sl;
            t1[u] = c1; t2[u] = c2; t3[u] = c3; t4[u] = c4;
        }
        a1 = __builtin_amdgcn_wmma_f32_16x16x4_f32(false, t1, false, bones, (short)0, a1, false, false);
        a2 = __builtin_amdgcn_wmma_f32_16x16x4_f32(false, t2, false, bones, (short)0, a2, false, false);
        a3 = __builtin_amdgcn_wmma_f32_16x16x4_f32(false, t3, false, bones, (short)0, a3, false, false);
        a4 = __builtin_amdgcn_wmma_f32_16x16x4_f32(false, t4, false, bones, (short)0, a4, false, false);
    }

    // All 16 columns of D are identical; rows 0-7 live on lanes 0-15, rows 8-15
    // on lanes 16-31.  Wave total = colsum(lane0) + colsum(lane16).
    float l1 = a1[0]+a1[1]+a1[2]+a1[3]+a1[4]+a1[5]+a1[6]+a1[7];
    float l2 = a2[0]+a2[1]+a2[2]+a2[3]+a2[4]+a2[5]+a2[6]+a2[7];
    float l3 = a3[0]+a3[1]+a3[2]+a3[3]+a3[4]+a3[5]+a3[6]+a3[7];
    float l4 = a4[0]+a4[1]+a4[2]+a4[3]+a4[4]+a4[5]+a4[6]+a4[7];
    float s1 = __shfl(l1, 0, 32) + __shfl(l1, 16, 32);
    float s2 = __shfl(l2, 0, 32) + __shfl(l2, 16, 32);
    float s3 = __shfl(l3, 0, 32) + __shfl(l3, 16, 32);
    float s4 = __shfl(l4, 0, 32) + __shfl(l4, 16, 32);

    __shared__ float wred[8][4];
    int wl = threadIdx.x >> 5;
    if (lane == 0) { wred[wl][0] = s1; wred[wl][1] = s2; wred[wl][2] = s3; wred[wl][3] = s4; }
    __syncthreads();
    if (threadIdx.x == 0) {
        float p0 = 0.f, p1 = 0.f, p2 = 0.f, p3 = 0.f;
        for (int w = 0; w < 8; ++w) { p0 += wred[w][0]; p1 += wred[w][1]; p2 += wred[w][2]; p3 += wred[w][3]; }
        partials[blockIdx.x * 4 + 0] = p0;
        partials[blockIdx.x * 4 + 1] = p1;
        partials[blockIdx.x * 4 + 2] = p2;
        partials[blockIdx.x * 4 + 3] = p3;
    }
}

// --------------------------------------------------------------- final scalar
__global__ void k_final(const float* __restrict__ partials, int nblocks, float* __restrict__ out) {
    if (threadIdx.x != 0 || blockIdx.x != 0) return;
    float S1 = 0.f, S2 = 0.f, S3 = 0.f, S4 = 0.f;
    for (int b = 0; b < nblocks; ++b) {
        S1 += partials[b * 4 + 0];
        S2 += partials[b * 4 + 1];
        S3 += partials[b * 4 + 2];
        S4 += partials[b * 4 + 3];
    }
    float wp = (S1 + 1.0f) / (S2 + 1.0f);        // SMOOTH = 1
    float wsn = (S3 + 1.0f) / (S4 + 1.0f);
    out[0] = 1.0f - 2.0f * (wp * wsn) / (wp + wsn);
}

extern "C" void kernel_launch(void* const* d_in, const int* in_sizes, int n_in,
                              void* d_out, int out_size, void* d_ws, size_t ws_size,
                              hipStream_t stream) {
    (void)in_sizes; (void)n_in; (void)out_size; (void)ws_size;
    const float* y_pred = (const float*)d_in[0];
    const int*   y_true = (const int*)d_in[1];
    float* out = (float*)d_out;
    float* ws = (float*)d_ws;

    float* mask = ws;                     // NV
    float* eA   = ws + (size_t)NV;        // NV
    float* eB   = ws + 2 * (size_t)NV;    // NV
    float* skel = ws + 3 * (size_t)NV;    // NV
    float* d2   = ws + 4 * (size_t)NV;    // NV
    float* rmax = ws + 5 * (size_t)NV;    // 4 (+pad)
    float* partials = rmax + 16;          // 256*4

    const int TB = 256;
    const int NB = (NV + TB - 1) / TB;

    k_binarize<<<(NBATCH + TB - 1) / TB, TB, 0, stream>>>(y_pred, y_true, mask);

    // Skeletonization: e_{k+1} = erode(e_k); init uses (mask, dilate(e_1));
    // iteration k uses (e_k, dilate(e_{k+1})).
    k_erode<<<NB, TB, 0, stream>>>(mask, eA);
    k_dilate_update<<<NB, TB, 0, stream>>>(eA, mask, skel, 1);
    float* xa = eA; float* xb = eB;
    for (int k = 0; k < 10; ++k) {
        k_erode<<<NB, TB, 0, stream>>>(xa, xb);
        k_dilate_update<<<NB, TB, 0, stream>>>(xb, xa, skel, 0);
        float* t = xa; xa = xb; xb = t;
    }

    // Exact EDT: init then three separable min-convolution passes (D, H, W).
    k_edt_init<<<NB, TB, 0, stream>>>(mask, d2);
    k_edt_pass<<<NVOL * Hh * Ww, 64, 0, stream>>>(d2, 64, HW, Hh, Ww, Ww, 1);  // axis D
    k_edt_pass<<<NVOL * Dd * Ww, 96, 0, stream>>>(d2, 96, Ww, Dd, HW, Ww, 1);  // axis H
    k_edt_pass<<<NVOL * Dd * Hh, 96, 0, stream>>>(d2, 96, 1, Dd, HW, Hh, Ww);  // axis W

    k_rmax_init<<<1, 32, 0, stream>>>(rmax);
    k_rmax<<<NVOL * 144, TB, 0, stream>>>(mask, skel, d2, rmax);

    const int SBLK = 256;
    k_sums_wmma<<<SBLK, 256, 0, stream>>>(mask, skel, d2, rmax, partials);
    k_final<<<1, 1, 0, stream>>>(partials, SBLK, out);
}